// GCN_60919816126943
// MI455X (gfx1250) — compile-verified
//
#include <hip/hip_runtime.h>

// ---------------------------------------------------------------------------
// GCN forward for gfx1250 (MI455X). f32 WMMA for the dense GEMMs, L2-resident
// atomic scatter for message passing.
// ---------------------------------------------------------------------------

#define NH 128           // NFEAT == NHID == 128
#define NCLASS 10
#define NGRAPH 512
#define LDA 132          // LDS row stride (floats), padded to dodge bank conflicts

typedef __attribute__((ext_vector_type(2))) float v2f;
typedef __attribute__((ext_vector_type(8))) float v8f;

// ---------------------------------------------------------------------------
// GEMM: H[M x 128] = X[M x 128] @ W[128 x 128], f32 WMMA 16x16x4.
// Block = 256 threads (8 waves). Block owns 16 rows; wave w owns cols 16w..16w+15.
// A tile (16x128) staged in LDS once; B slab preloaded to 64 VGPRs per wave.
// M must be a multiple of 16 (50000 = 3125 * 16).
// ---------------------------------------------------------------------------
__global__ __launch_bounds__(256)
void gcn_gemm_wmma(const float* __restrict__ X, const float* __restrict__ W,
                   float* __restrict__ H, int M) {
    __shared__ float lA[16 * LDA];
    const int row0 = blockIdx.x * 16;
    const int tid  = threadIdx.x;

    // Cooperative A-tile load: 16x128 floats, 8 floats (2x float4) per thread.
    {
        const int r = tid >> 4;            // 0..15
        const int c = (tid & 15) * 8;      // 0..120
        const float4* src = (const float4*)(X + (size_t)(row0 + r) * NH + c);
        float4 a0 = src[0];
        float4 a1 = src[1];
        float4* dst = (float4*)&lA[r * LDA + c];  // 16B aligned: (r*528 + c*4) % 16 == 0
        dst[0] = a0;
        dst[1] = a1;
    }
    __syncthreads();

    const int lane = tid & 31;
    const int wave = tid >> 5;       // 0..7
    const int half = lane >> 4;      // 0 or 1
    const int lm   = lane & 15;      // 0..15
    const int col0 = wave * 16;

    // Preload B slab: b[s] holds W[4s+2*half + {0,1}][col0+lm]  (32 v2f = 64 VGPRs)
    v2f b[32];
#pragma unroll
    for (int s = 0; s < 32; ++s) {
        const int k = 4 * s + 2 * half;
        b[s][0] = W[(size_t)k * NH + col0 + lm];
        b[s][1] = W[(size_t)(k + 1) * NH + col0 + lm];
    }

    v8f acc = {};
#pragma unroll
    for (int s = 0; s < 32; ++s) {
        const int k = 4 * s + 2 * half;
        v2f a = *(const v2f*)&lA[lm * LDA + k];   // 8B aligned (k even, LDA even)
        // D = A x B + C   (16x16x4, f32)
        acc = __builtin_amdgcn_wmma_f32_16x16x4_f32(
            /*neg_a=*/false, a, /*neg_b=*/false, b[s],
            /*c_mod=*/(short)0, acc, /*reuse_a=*/false, /*reuse_b=*/false);
    }

    // D layout: acc[r] = D[r + 8*half][lm]
#pragma unroll
    for (int r = 0; r < 8; ++r) {
        H[(size_t)(row0 + r + 8 * half) * NH + col0 + lm] = acc[r];
    }
}

// ---------------------------------------------------------------------------
// Degree / normalization
// ---------------------------------------------------------------------------
__global__ void deg_init(float* __restrict__ deg, int n) {
    int i = blockIdx.x * blockDim.x + threadIdx.x;
    if (i < n) deg[i] = 1.0f;   // self-loop contributes 1 to every node
}

__global__ void deg_edges(const int* __restrict__ dst, float* __restrict__ deg, int E) {
    int i = blockIdx.x * blockDim.x + threadIdx.x;
    if (i < E) atomicAdd(&deg[dst[i]], 1.0f);
}

__global__ void deg_to_dis(float* __restrict__ deg, int n) {
    int i = blockIdx.x * blockDim.x + threadIdx.x;
    if (i < n) deg[i] = rsqrtf(deg[i]);   // deg >= 1 always (self-loops)
}

// ---------------------------------------------------------------------------
// out[i,:] = hw[i,:] * dis[i]^2   (self-loop term; fully initializes out)
// One wave per node; lane handles 4 feats (float4).
// ---------------------------------------------------------------------------
__global__ void selfloop_init(const float* __restrict__ hw, const float* __restrict__ dis,
                              float* __restrict__ out, int n) {
    int t = blockIdx.x * blockDim.x + threadIdx.x;
    int node = t >> 5;
    int f = (t & 31) * 4;
    if (node < n) {
        float w = dis[node];
        w = w * w;
        float4 v = *(const float4*)(hw + (size_t)node * NH + f);
        float4 o;
        o.x = v.x * w; o.y = v.y * w; o.z = v.z * w; o.w = v.w * w;
        *(float4*)(out + (size_t)node * NH + f) = o;
    }
}

// ---------------------------------------------------------------------------
// Edge scatter: out[dst,:] += hw[src,:] * dis[src]*dis[dst]
// One wave per edge; lane handles 4 feats; 4 f32 atomics (L2-resident).
// ---------------------------------------------------------------------------
__global__ void edge_scatter(const float* __restrict__ hw, const int* __restrict__ src,
                             const int* __restrict__ dst, const float* __restrict__ dis,
                             float* __restrict__ out, int E) {
    int t = blockIdx.x * blockDim.x + threadIdx.x;
    int e = t >> 5;
    int f = (t & 31) * 4;
    if (e < E) {
        int s = src[e];
        int d = dst[e];
        float w = dis[s] * dis[d];
        float4 v = *(const float4*)(hw + (size_t)s * NH + f);
        float* o = out + (size_t)d * NH + f;
        atomicAdd(o + 0, v.x * w);
        atomicAdd(o + 1, v.y * w);
        atomicAdd(o + 2, v.z * w);
        atomicAdd(o + 3, v.w * w);
    }
}

// ---------------------------------------------------------------------------
// h = relu(out + b)   (in place allowed)
// ---------------------------------------------------------------------------
__global__ void bias_relu(const float* __restrict__ out, const float* __restrict__ b,
                          float* __restrict__ h, int total) {
    int t = blockIdx.x * blockDim.x + threadIdx.x;
    if (t < total) {
        float v = out[t] + b[t & (NH - 1)];
        h[t] = v > 0.0f ? v : 0.0f;
    }
}

// ---------------------------------------------------------------------------
// Pooling accumulators
// ---------------------------------------------------------------------------
__global__ void zero_f32(float* __restrict__ p, int n) {
    int i = blockIdx.x * blockDim.x + threadIdx.x;
    if (i < n) p[i] = 0.0f;
}

__global__ void pool_accum(const float* __restrict__ h, const int* __restrict__ batch,
                           float* __restrict__ psum, float* __restrict__ pcnt, int n) {
    int t = blockIdx.x * blockDim.x + threadIdx.x;
    int node = t >> 5;
    int lane = t & 31;
    int f = lane * 4;
    if (node < n) {
        int g = batch[node];
        float4 v = *(const float4*)(h + (size_t)node * NH + f);
        float* p = psum + (size_t)g * NH + f;
        atomicAdd(p + 0, v.x);
        atomicAdd(p + 1, v.y);
        atomicAdd(p + 2, v.z);
        atomicAdd(p + 3, v.w);
        if (lane == 0) atomicAdd(&pcnt[g], 1.0f);
    }
}

// ---------------------------------------------------------------------------
// Head: pooled = psum/cnt; logits = pooled @ lin_w + lin_b; log_softmax.
// One thread per graph (512 threads total; trivial work).
// ---------------------------------------------------------------------------
__global__ void head_kernel(const float* __restrict__ psum, const float* __restrict__ pcnt,
                            const float* __restrict__ lin_w, const float* __restrict__ lin_b,
                            float* __restrict__ out, int G) {
    int g = blockIdx.x * blockDim.x + threadIdx.x;
    if (g >= G) return;
    float inv = 1.0f / fmaxf(pcnt[g], 1.0f);
    float logit[NCLASS];
#pragma unroll
    for (int c = 0; c < NCLASS; ++c) logit[c] = lin_b[c];
    const float* row = psum + (size_t)g * NH;
    for (int k = 0; k < NH; ++k) {
        float xv = row[k] * inv;
        const float* wr = lin_w + (size_t)k * NCLASS;
#pragma unroll
        for (int c = 0; c < NCLASS; ++c) logit[c] = fmaf(xv, wr[c], logit[c]);
    }
    float m = logit[0];
#pragma unroll
    for (int c = 1; c < NCLASS; ++c) m = fmaxf(m, logit[c]);
    float se = 0.0f;
#pragma unroll
    for (int c = 0; c < NCLASS; ++c) se += __expf(logit[c] - m);
    float lse = m + __logf(se);
    float* o = out + (size_t)g * NCLASS;
#pragma unroll
    for (int c = 0; c < NCLASS; ++c) o[c] = logit[c] - lse;
}

// ---------------------------------------------------------------------------
// Launch
// ---------------------------------------------------------------------------
extern "C" void kernel_launch(void* const* d_in, const int* in_sizes, int n_in,
                              void* d_out, int out_size, void* d_ws, size_t ws_size,
                              hipStream_t stream) {
    const float* x     = (const float*)d_in[0];
    const int*   eidx  = (const int*)d_in[1];      // [2, E]
    const int*   batch = (const int*)d_in[2];
    const float* W0 = (const float*)d_in[3];
    const float* b0 = (const float*)d_in[4];
    const float* W1 = (const float*)d_in[5];
    const float* b1 = (const float*)d_in[6];
    const float* W2 = (const float*)d_in[7];
    const float* b2 = (const float*)d_in[8];
    const float* lw = (const float*)d_in[9];
    const float* lb = (const float*)d_in[10];
    float* out = (float*)d_out;

    const int N = in_sizes[0] / NH;        // 50000
    const int E = in_sizes[1] / 2;         // 800000
    const int* src = eidx;
    const int* dst = eidx + E;

    // Workspace layout
    char* ws = (char*)d_ws;
    const size_t featBytes = (size_t)N * NH * sizeof(float);
    float* bufA = (float*)ws;                                // hw (GEMM output)
    float* bufB = (float*)(ws + featBytes);                  // accum / h
    float* dis  = (float*)(ws + 2 * featBytes);              // N floats (deg -> dis)
    float* psum = (float*)(ws + 2 * featBytes + ((size_t)N * sizeof(float) + 255 & ~size_t(255)));
    float* pcnt = psum + (size_t)NGRAPH * NH;

    const int TB = 256;
    auto blocks = [](long long t, int b) { return (int)((t + b - 1) / b); };

    // --- normalization ---
    deg_init<<<blocks(N, TB), TB, 0, stream>>>(dis, N);
    deg_edges<<<blocks(E, TB), TB, 0, stream>>>(dst, dis, E);
    deg_to_dis<<<blocks(N, TB), TB, 0, stream>>>(dis, N);

    const int gemmBlocks = N / 16;                     // 3125 (N multiple of 16)
    const long long nodeLanes = (long long)N * 32;
    const long long edgeLanes = (long long)E * 32;

    const float* hin = x;
    const float* Ws[3] = {W0, W1, W2};
    const float* bs[3] = {b0, b1, b2};
    for (int layer = 0; layer < 3; ++layer) {
        gcn_gemm_wmma<<<gemmBlocks, TB, 0, stream>>>(hin, Ws[layer], bufA, N);
        selfloop_init<<<blocks(nodeLanes, TB), TB, 0, stream>>>(bufA, dis, bufB, N);
        edge_scatter<<<blocks(edgeLanes, TB), TB, 0, stream>>>(bufA, src, dst, dis, bufB, E);
        bias_relu<<<blocks((long long)N * NH, TB), TB, 0, stream>>>(bufB, bs[layer], bufB, N * NH);
        hin = bufB;   // next layer reads h; GEMM writes bufA, then accum reuses bufB
    }

    // --- pooling + head ---
    zero_f32<<<blocks(NGRAPH * NH + NGRAPH, TB), TB, 0, stream>>>(psum, NGRAPH * NH + NGRAPH);
    pool_accum<<<blocks(nodeLanes, TB), TB, 0, stream>>>(bufB, batch, psum, pcnt, N);
    head_kernel<<<blocks(NGRAPH, TB), TB, 0, stream>>>(psum, pcnt, lw, lb, out, NGRAPH);
}